// Group_Attention_26465588478181
// MI455X (gfx1250) — compile-verified
//
#include <hip/hip_runtime.h>
#include <math.h>

#define Bn 2
#define Cn 96
#define Nn 1000
#define Wn 96
#define Gn 6
#define DG 16
#define NW (Nn*Wn)            /* 96000  */
#define TOT (Bn*Cn*Nn*Wn)     /* 18432000 */
#define EPSN 1e-5f

typedef float v2f __attribute__((ext_vector_type(2)));
typedef float v8f __attribute__((ext_vector_type(8)));

__device__ __forceinline__ v8f wmma4(v2f a, v2f b, v8f c) {
  // D = A(16x4 f32) * B(4x16 f32) + C(16x16 f32)
  return __builtin_amdgcn_wmma_f32_16x16x4_f32(false, a, false, b, (short)0, c, false, false);
}

// ---------------------------------------------------------------------------
// Zero small stats region (also provides the always-valid zero-bias vector)
// ---------------------------------------------------------------------------
__global__ void zero_stats_k(float* p, int n) {
  for (int i = threadIdx.x; i < n; i += blockDim.x) p[i] = 0.f;
}

// ---------------------------------------------------------------------------
// 96-out-channel 1x1 conv as WMMA f32 GEMM:  Y[ch][j] = sum_k Wm[ch][k]*X[k][j]
// X,Y layout [B,96,N,W] flattened; j enumerates (b,n,w). One 16x16 tile/wave.
// bias is always a valid pointer (zero vector when unused) -> branch-free
// epilogue. Optional fused per-(b,ch) sum/sumsq atomics for IN/BN statistics.
// ---------------------------------------------------------------------------
__global__ void gemm96_wmma_k(const float* __restrict__ X,
                              const float* __restrict__ Wm,
                              const float* __restrict__ bias,
                              float* __restrict__ Y,
                              float* __restrict__ inst_sum,
                              float* __restrict__ inst_sq) {
  const int wave = (int)((blockIdx.x * blockDim.x + threadIdx.x) >> 5);
  const int lane = threadIdx.x & 31;
  const int mt = wave % 6;
  const long jt = wave / 6;
  const long Jtiles = (long)Bn * NW / 16;
  if (jt >= Jtiles) return;                       // wave-uniform
  const long j0 = jt * 16;
  const int b = (int)(j0 / NW);
  const long base = j0 + (long)b * (Cn - 1) * NW; // flat offset of (b, k=0, j0-local)
  const int lh = lane >> 4;                       // half-wave select
  const int ln = lane & 15;
  const int m0 = mt * 16;

  // branch-free per-lane bias for the 8 accumulator channels
  float bch[8];
#pragma unroll
  for (int v = 0; v < 8; ++v) bch[v] = bias[m0 + v + 8 * lh];

  v8f acc = {0.f,0.f,0.f,0.f,0.f,0.f,0.f,0.f};
  const float* wrow = Wm + (m0 + ln) * Cn;        // A row for this lane (M = ln)
  for (int k0 = 0; k0 < Cn; k0 += 4) {
    v2f a, bv;
    a.x = wrow[k0 + 2 * lh];                      // A: K = 2*lh + vgpr
    a.y = wrow[k0 + 2 * lh + 1];
    const float* xp = X + base + (long)(k0 + 2 * lh) * NW + ln;  // B: K = 2*lh + vgpr, N = ln
    bv.x = xp[0];
    bv.y = xp[NW];
    acc = wmma4(a, bv, acc);
  }

  float vsum[8], vsq[8];
#pragma unroll
  for (int v = 0; v < 8; ++v) {
    const int ch = m0 + v + 8 * lh;               // C layout: M = v + 8*lh, N = ln
    const float val = acc[v] + bch[v];
    Y[base + (long)ch * NW + ln] = val;
    vsum[v] = val; vsq[v] = val * val;
  }
  if (inst_sum) {                                 // wave-uniform branch
#pragma unroll
    for (int v = 0; v < 8; ++v) {
      float s = vsum[v], q = vsq[v];
      for (int off = 1; off < 16; off <<= 1) {
        s += __shfl_xor(s, off, 16);
        q += __shfl_xor(q, off, 16);
      }
      if (ln == 0) {
        const int ch = m0 + v + 8 * lh;
        atomicAdd(&inst_sum[b * Cn + ch], s);
        atomicAdd(&inst_sq[b * Cn + ch], q);
      }
    }
  }
}

// ---------------------------------------------------------------------------
// Per-head block means -> logits -> Sinkhorn(8) -> exp -> top-3 cut -> P
// One 192-thread block per head (h = (b*96 + c)*6 + g), 1152 heads.
// ---------------------------------------------------------------------------
__global__ void sinkhorn_P_k(const float* __restrict__ feat,
                             float* __restrict__ P, float invtemp) {
  const int h = blockIdx.x;
  const int g = h % Gn;
  const int c = (h / Gn) % Cn;
  const int b = h / (Gn * Cn);
  const float* fb = feat + ((long)(b * Cn + c) * Nn) * Wn + g * DG;

  __shared__ float km[10 * 16];
  __shared__ float L[100];
  __shared__ float red[10];
  const int t = threadIdx.x;

  if (t < 160) {                                  // block means (q==k==f)
    const int blk = t >> 4, dim = t & 15;
    float s = 0.f;
    for (int i = 0; i < 100; ++i) s += fb[(long)(blk * 100 + i) * Wn + dim];
    km[t] = s * 0.01f;
  }
  __syncthreads();
  if (t < 100) {
    const int i = t / 10, j = t % 10;
    float s = 0.f;
    for (int d = 0; d < 16; ++d) s += km[i * 16 + d] * km[j * 16 + d];
    L[t] = s * invtemp;
  }
  __syncthreads();
  for (int it = 0; it < 8; ++it) {
    if (t < 10) {                                 // row logsumexp
      float mx = -1e30f;
      for (int j = 0; j < 10; ++j) mx = fmaxf(mx, L[t * 10 + j]);
      float s = 0.f;
      for (int j = 0; j < 10; ++j) s += expf(L[t * 10 + j] - mx);
      red[t] = mx + logf(s);
    }
    __syncthreads();
    if (t < 100) L[t] -= red[t / 10];
    __syncthreads();
    if (t < 10) {                                 // col logsumexp
      float mx = -1e30f;
      for (int i = 0; i < 10; ++i) mx = fmaxf(mx, L[i * 10 + t]);
      float s = 0.f;
      for (int i = 0; i < 10; ++i) s += expf(L[i * 10 + t] - mx);
      red[t] = mx + logf(s);
    }
    __syncthreads();
    if (t < 100) L[t] -= red[t % 10];
    __syncthreads();
  }
  if (t < 100) L[t] = expf(L[t]);
  __syncthreads();
  if (t < 10) {                                   // 3rd largest per row
    float t0 = -1e30f, t1 = -1e30f, t2 = -1e30f;
    for (int j = 0; j < 10; ++j) {
      float v = L[t * 10 + j];
      if (v > t0) { t2 = t1; t1 = t0; t0 = v; }
      else if (v > t1) { t2 = t1; t1 = v; }
      else if (v > t2) { t2 = v; }
    }
    red[t] = t2;
  }
  __syncthreads();
  if (t < 100) {
    const float v = L[t];
    P[(long)h * 100 + t] = (v >= red[t / 10]) ? v : 0.f;
  }
}

// ---------------------------------------------------------------------------
// Attention: one 256-thread WG per (head, query-block). WMMA f32 score GEMM
// (112-padded), softmax, WMMA apply GEMM. LDS ~60.4 KB.
// ---------------------------------------------------------------------------
__global__ void attn_wmma_k(const float* __restrict__ feat,
                            const float* __restrict__ P,
                            float* __restrict__ attn, float invtemp) {
  __shared__ float sk_s[112 * 17];
  __shared__ float sv_s[112 * 17];
  __shared__ float S_s[100 * 113];

  const int task = blockIdx.x;
  const int m = task % 10;
  const int h = task / 10;
  const int g = h % Gn;
  const int c = (h / Gn) % Cn;
  const int b = h / (Gn * Cn);
  const float* fb = feat + ((long)(b * Cn + c) * Nn) * Wn + g * DG;
  float* ob = attn + ((long)(b * Cn + c) * Nn) * Wn + g * DG;
  const float* Ph = P + (long)h * 100 + m * 10;
  const int tid = threadIdx.x;

  // sk = P_row . kb ; sv = P_row . elu(kb)  (rows >= 100 zero-padded)
  for (int idx = tid; idx < 112 * 16; idx += 256) {
    const int t = idx >> 4, d = idx & 15;
    float aK = 0.f, aV = 0.f;
    if (t < 100) {
      for (int n = 0; n < 10; ++n) {
        const float p = Ph[n];
        if (p != 0.f) {
          const float kv = fb[(long)(n * 100 + t) * Wn + d];
          const float vv = kv > 0.f ? kv : expm1f(kv);
          aK += p * kv; aV += p * vv;
        }
      }
    }
    sk_s[t * 17 + d] = aK;
    sv_s[t * 17 + d] = aV;
  }
  __syncthreads();

  const int wave = tid >> 5, lane = tid & 31, lh = lane >> 4, ln = lane & 15;

  if (wave < 7) {                                 // S = qb . sk^T  (K = 16)
    const int mt = wave;
    int s = mt * 16 + ln; if (s > 99) s = 99;     // clamp padded query rows
    const float* qp = fb + (long)(m * 100 + s) * Wn;
    v2f a_reg[4];
#pragma unroll
    for (int kk = 0; kk < 4; ++kk) {              // A: K = 4*kk + 2*lh + vgpr
      a_reg[kk].x = qp[4 * kk + 2 * lh];
      a_reg[kk].y = qp[4 * kk + 2 * lh + 1];
    }
    for (int nt = 0; nt < 7; ++nt) {
      v8f acc = {0.f,0.f,0.f,0.f,0.f,0.f,0.f,0.f};
#pragma unroll
      for (int kk = 0; kk < 4; ++kk) {
        v2f bv;                                   // B[k][n] = sk[n][k]
        bv.x = sk_s[(nt * 16 + ln) * 17 + 4 * kk + 2 * lh];
        bv.y = sk_s[(nt * 16 + ln) * 17 + 4 * kk + 2 * lh + 1];
        acc = wmma4(a_reg[kk], bv, acc);
      }
#pragma unroll
      for (int v = 0; v < 8; ++v) {
        const int r = mt * 16 + v + 8 * lh;
        if (r < 100) S_s[r * 113 + nt * 16 + ln] = acc[v];
      }
    }
  }
  __syncthreads();

  for (int r = tid; r < 100; r += 256) {          // softmax over 100 keys
    float mx = -1e30f;
    for (int j = 0; j < 100; ++j) mx = fmaxf(mx, S_s[r * 113 + j] * invtemp);
    float sum = 0.f;
    for (int j = 0; j < 100; ++j) {
      const float e = expf(S_s[r * 113 + j] * invtemp - mx);
      S_s[r * 113 + j] = e; sum += e;
    }
    const float inv = 1.f / sum;
    for (int j = 0; j < 100; ++j) S_s[r * 113 + j] *= inv;
    for (int j = 100; j < 113; ++j) S_s[r * 113 + j] = 0.f;  // zero K-padding
  }
  __syncthreads();

  if (wave < 7) {                                 // O = A . sv   (K = 112)
    const int mt = wave;
    int r = mt * 16 + ln; if (r > 99) r = 99;
    v8f acc = {0.f,0.f,0.f,0.f,0.f,0.f,0.f,0.f};
    for (int k0 = 0; k0 < 112; k0 += 4) {
      v2f a, bv;
      a.x = S_s[r * 113 + k0 + 2 * lh];
      a.y = S_s[r * 113 + k0 + 2 * lh + 1];
      bv.x = sv_s[(k0 + 2 * lh) * 17 + ln];
      bv.y = sv_s[(k0 + 2 * lh + 1) * 17 + ln];
      acc = wmma4(a, bv, acc);
    }
#pragma unroll
    for (int v = 0; v < 8; ++v) {
      const int s = mt * 16 + v + 8 * lh;
      if (s < 100) ob[(long)(m * 100 + s) * Wn + ln] = acc[v];
    }
  }
}

// ---------------------------------------------------------------------------
// GroupNorm of (attn^T(1,3) + x): reduction per (b, group-of-new-channel)
// ---------------------------------------------------------------------------
__global__ void gn_reduce_k(const float* __restrict__ attn,
                            const float* __restrict__ x,
                            float* __restrict__ gsum, float* __restrict__ gsq) {
  const int bc = blockIdx.x;                      // (b, cy), 192 blocks
  const int b = bc / Cn, cy = bc % Cn;
  const float* xrow = x + (long)(b * Cn + cy) * NW;
  float s = 0.f, q = 0.f;
  for (int i = threadIdx.x; i < NW; i += 256) {
    const int n = i / Wn, wy = i % Wn;
    const float pre = attn[((long)(b * Cn + wy) * Nn + n) * Wn + cy] + xrow[i];
    s += pre; q += pre * pre;
  }
  __shared__ float rs[256], rq[256];
  rs[threadIdx.x] = s; rq[threadIdx.x] = q;
  __syncthreads();
  for (int off = 128; off > 0; off >>= 1) {
    if (threadIdx.x < off) { rs[threadIdx.x] += rs[threadIdx.x + off];
                             rq[threadIdx.x] += rq[threadIdx.x + off]; }
    __syncthreads();
  }
  if (threadIdx.x == 0) {
    const int gi = b * Gn + cy / DG;
    atomicAdd(&gsum[gi], rs[0]);
    atomicAdd(&gsq[gi], rq[0]);
  }
}

__global__ void gn_apply_k(const float* __restrict__ attn,
                           const float* __restrict__ x,
                           const float* __restrict__ gsum,
                           const float* __restrict__ gsq,
                           const float* __restrict__ gw,
                           const float* __restrict__ gb,
                           float* __restrict__ y) {
  const float cnt = (float)DG * (float)NW;        // 1,536,000
  for (long i = blockIdx.x * (long)blockDim.x + threadIdx.x; i < TOT;
       i += (long)gridDim.x * blockDim.x) {
    const int wy = (int)(i % Wn);
    const int n  = (int)((i / Wn) % Nn);
    const int cy = (int)((i / NW) % Cn);
    const int b  = (int)(i / ((long)Cn * NW));
    const float pre = attn[((long)(b * Cn + wy) * Nn + n) * Wn + cy] + x[i];
    const int gi = b * Gn + cy / DG;
    const float m = gsum[gi] / cnt;
    const float v = gsq[gi] / cnt - m * m;
    y[i] = (pre - m) * rsqrtf(v + EPSN) * gw[cy] + gb[cy];
  }
}

// ---------------------------------------------------------------------------
// left-branch 1: relu(BN(IN(conv))) with BN of IN folded analytically
// ---------------------------------------------------------------------------
__global__ void in_bn_relu_k(float* __restrict__ t,
                             const float* __restrict__ isum,
                             const float* __restrict__ isq,
                             const float* __restrict__ bnw,
                             const float* __restrict__ bnb) {
  const float cnt = (float)NW;
  for (long i = blockIdx.x * (long)blockDim.x + threadIdx.x; i < TOT;
       i += (long)gridDim.x * blockDim.x) {
    const int ch = (int)((i / NW) % Cn);
    const int b  = (int)(i / ((long)Cn * NW));
    const float m0 = isum[0 * Cn + ch] / cnt, m1 = isum[1 * Cn + ch] / cnt;
    const float v0 = isq[0 * Cn + ch] / cnt - m0 * m0;
    const float v1 = isq[1 * Cn + ch] / cnt - m1 * m1;
    const float varz = 0.5f * (v0 / (v0 + EPSN) + v1 / (v1 + EPSN));
    const float mb = (b == 0) ? m0 : m1;
    const float vb = (b == 0) ? v0 : v1;
    const float sc = rsqrtf(vb + EPSN) * rsqrtf(varz + EPSN) * bnw[ch];
    const float o = (t[i] - mb) * sc + bnb[ch];
    t[i] = o > 0.f ? o : 0.f;
  }
}

// ---------------------------------------------------------------------------
// final: relu( BN(IN(conv_l2)) + BN(conv_right) )
// ---------------------------------------------------------------------------
__global__ void final_k(const float* __restrict__ tl2,
                        const float* __restrict__ isumL, const float* __restrict__ isqL,
                        const float* __restrict__ bn2w, const float* __restrict__ bn2b,
                        const float* __restrict__ tr,
                        const float* __restrict__ isumR, const float* __restrict__ isqR,
                        const float* __restrict__ bnrw, const float* __restrict__ bnrb,
                        float* __restrict__ out) {
  const float cnt = (float)NW;
  const float cntA = 2.f * (float)NW;
  for (long i = blockIdx.x * (long)blockDim.x + threadIdx.x; i < TOT;
       i += (long)gridDim.x * blockDim.x) {
    const int ch = (int)((i / NW) % Cn);
    const int b  = (int)(i / ((long)Cn * NW));
    // left: IN then analytic BN
    const float m0 = isumL[0 * Cn + ch] / cnt, m1 = isumL[1 * Cn + ch] / cnt;
    const float v0 = isqL[0 * Cn + ch] / cnt - m0 * m0;
    const float v1 = isqL[1 * Cn + ch] / cnt - m1 * m1;
    const float varz = 0.5f * (v0 / (v0 + EPSN) + v1 / (v1 + EPSN));
    const float mb = (b == 0) ? m0 : m1;
    const float vb = (b == 0) ? v0 : v1;
    const float left = (tl2[i] - mb) * rsqrtf(vb + EPSN) * rsqrtf(varz + EPSN) * bn2w[ch] + bn2b[ch];
    // right: plain BN over (B,H,W)
    const float mr = (isumR[ch] + isumR[Cn + ch]) / cntA;
    const float vr = (isqR[ch] + isqR[Cn + ch]) / cntA - mr * mr;
    const float right = (tr[i] - mr) * rsqrtf(vr + EPSN) * bnrw[ch] + bnrb[ch];
    const float o = left + right;
    out[i] = o > 0.f ? o : 0.f;
  }
}

// ---------------------------------------------------------------------------
extern "C" void kernel_launch(void* const* d_in, const int* in_sizes, int n_in,
                              void* d_out, int out_size, void* d_ws, size_t ws_size,
                              hipStream_t stream) {
  const float* x       = (const float*)d_in[0];
  const float* w_lin   = (const float*)d_in[1];
  const float* gn_w    = (const float*)d_in[2];
  const float* gn_b    = (const float*)d_in[3];
  const float* w_right = (const float*)d_in[4];
  const float* b_right = (const float*)d_in[5];
  const float* bn_r_w  = (const float*)d_in[6];
  const float* bn_r_b  = (const float*)d_in[7];
  const float* w_l1    = (const float*)d_in[8];
  const float* b_l1    = (const float*)d_in[9];
  const float* bn1_w   = (const float*)d_in[10];
  const float* bn1_b   = (const float*)d_in[11];
  const float* w_l2    = (const float*)d_in[12];
  const float* b_l2    = (const float*)d_in[13];
  const float* bn2_w   = (const float*)d_in[14];
  const float* bn2_b   = (const float*)d_in[15];
  float* out = (float*)d_out;

  float* ws = (float*)d_ws;
  const long SB = (long)TOT;
  float* big0 = ws;                // feat -> t_l1 (in-place relu-norm)
  float* big1 = ws + SB;           // attn -> t_right
  float* big2 = ws + 2 * SB;       // y    -> t_l2
  float* Pbuf = ws + 3 * SB;       // 1152*100
  float* stats = Pbuf + 1152 * 100;
  float* gsum  = stats;            // 12
  float* gsq   = stats + 12;       // 12
  float* isumR = stats + 24;       // 192
  float* isqR  = isumR + 192;
  float* isumL1 = isqR + 192;
  float* isqL1  = isumL1 + 192;
  float* isumL2 = isqL1 + 192;
  float* isqL2  = isumL2 + 192;
  float* zbias  = isqL2 + 192;     // 96-float always-zero bias vector
  const int n_stats = 24 + 6 * 192 + 96;

  const float invtemp = 1.0f / sqrtf((float)Cn);
  const int gemm_blocks = 9000;    // 72000 waves = 6 Mtiles x 12000 Jtiles

  zero_stats_k<<<1, 256, 0, stream>>>(stats, n_stats);

  // feat = W_lin @ x
  gemm96_wmma_k<<<gemm_blocks, 256, 0, stream>>>(x, w_lin, zbias, big0, nullptr, nullptr);
  // Sinkhorn transport plan
  sinkhorn_P_k<<<1152, 192, 0, stream>>>(big0, Pbuf, invtemp);
  // blocked attention
  attn_wmma_k<<<11520, 256, 0, stream>>>(big0, Pbuf, big1, invtemp);
  // y = GN(attn^T + x)
  gn_reduce_k<<<192, 256, 0, stream>>>(big1, x, gsum, gsq);
  gn_apply_k<<<4608, 256, 0, stream>>>(big1, x, gsum, gsq, gn_w, gn_b, big2);
  // ResNet block
  gemm96_wmma_k<<<gemm_blocks, 256, 0, stream>>>(big2, w_right, b_right, big1, isumR, isqR);
  gemm96_wmma_k<<<gemm_blocks, 256, 0, stream>>>(big2, w_l1, b_l1, big0, isumL1, isqL1);
  in_bn_relu_k<<<4608, 256, 0, stream>>>(big0, isumL1, isqL1, bn1_w, bn1_b);
  gemm96_wmma_k<<<gemm_blocks, 256, 0, stream>>>(big0, w_l2, b_l2, big2, isumL2, isqL2);
  final_k<<<4608, 256, 0, stream>>>(big2, isumL2, isqL2, bn2_w, bn2_b,
                                    big1, isumR, isqR, bn_r_w, bn_r_b, out);
}